// ExponentialMovingAverage_1451698946750
// MI455X (gfx1250) — compile-verified
//
#include <hip/hip_runtime.h>
#include <hip/hip_bf16.h>
#include <cstdint>

// ---------------------------------------------------------------------------
// EMA final-state:  out[b,d] = sum_t 0.1 * 0.9^(S-1-t) * x[b,t,d]
//   x: [8, 4096, 1024] f32   out: [8, 1024] f32
// Pass 1: per-(batch,chunk) local EMA partials, async-LDS double-buffered
//         streaming (bandwidth-bound: 128 MiB @ 23.3 TB/s ~ 5.5 us).
// Pass 2: geometric combine of 64 chunk partials via V_WMMA_F32_16X16X4_F32.
// ---------------------------------------------------------------------------

#define EMA_ALPHA   0.1f
#define EMA_OMA     0.9f
#define BATCH       8
#define SEQ         4096
#define DIM         1024
#define NCHUNK      64
#define CHUNKLEN    64                       // SEQ / NCHUNK
#define ROWS_PER_G  8
#define NGROUP      (CHUNKLEN / ROWS_PER_G)  // 8 groups of 8 rows
#define ROW_BYTES   (DIM * 4)                // 4096 B per timestep row
#define GROUP_BYTES (ROWS_PER_G * ROW_BYTES) // 32768 B per stage
#define STAGE_F     (ROWS_PER_G * DIM)       // 8192 floats per stage

typedef __attribute__((ext_vector_type(2))) float v2f;
typedef __attribute__((ext_vector_type(8))) float v8f;

// Issue one stage (8 rows x 4 KB) of async global->LDS copies.
// Per CDNA5 ISA, INST_OFFSET is added to BOTH the LDS address
// (VDST VGPR) and the global address, so one immediate advances both sides
// by the row stride. Tracked on ASYNCcnt.
// `loff` carries the ptrtoint'd LDS address of the __shared__ buffer, which
// both supplies the hardware LDS offset (flat addr low 32 bits == wave-
// relative LDS offset) and CAPTURES the buffer so the "memory" clobber is
// known to alias it (otherwise LLVM dead-codes the LDS read-back).
__device__ __forceinline__ void async_prefetch_group(const float* gbase,
                                                     uint32_t goff,
                                                     uint32_t loff) {
  asm volatile(
      "global_load_async_to_lds_b128 %0, %1, %2 offset:0\n\t"
      "global_load_async_to_lds_b128 %0, %1, %2 offset:4096\n\t"
      "global_load_async_to_lds_b128 %0, %1, %2 offset:8192\n\t"
      "global_load_async_to_lds_b128 %0, %1, %2 offset:12288\n\t"
      "global_load_async_to_lds_b128 %0, %1, %2 offset:16384\n\t"
      "global_load_async_to_lds_b128 %0, %1, %2 offset:20480\n\t"
      "global_load_async_to_lds_b128 %0, %1, %2 offset:24576\n\t"
      "global_load_async_to_lds_b128 %0, %1, %2 offset:28672"
      :: "v"(loff), "v"(goff), "s"(gbase)
      : "memory");
}

// Pass 1: block = one (batch, chunk); 256 threads, each owns 4 dims.
// Local recurrence acc = 0.9*acc + 0.1*x gives the chunk partial with
// relative weights 0.1*0.9^(L-1-t); absolute chunk scale applied in pass 2.
__global__ __launch_bounds__(256) void ema_chunk_partial(
    const float* __restrict__ x, float* __restrict__ ws) {
  __shared__ float smem[2 * STAGE_F];  // 64 KB double buffer (WGP has 320 KB)

  const uint32_t tid  = threadIdx.x;
  const uint32_t boff = tid * 16u;     // this thread's 16 B within a row

  // Generic (flat) address of LDS is SHARED_BASE|offset; low 32 bits are the
  // wave-relative LDS byte offset the async engine expects in VDST. The
  // ptrtoint also escapes `smem` -> asm memory clobbers may write it.
  const uint32_t lds0 = (uint32_t)(uintptr_t)(void*)&smem[0];

  const float* base = x + (size_t)blockIdx.x * (size_t)(CHUNKLEN * DIM);

  // Prime stage 0.
  async_prefetch_group(base, boff, lds0 + boff);

  float a0 = 0.f, a1 = 0.f, a2 = 0.f, a3 = 0.f;

  for (int g = 0; g < NGROUP; ++g) {
    if (g + 1 < NGROUP) {
      // Prefetch next stage, then wait for the *previous* 8 loads only
      // (async loads complete in order -> asynccnt<=8 means stage g landed).
      async_prefetch_group(base,
                           (uint32_t)(g + 1) * GROUP_BYTES + boff,
                           lds0 + (uint32_t)((g + 1) & 1) * GROUP_BYTES + boff);
      asm volatile("s_wait_asynccnt 8" ::: "memory");
    } else {
      asm volatile("s_wait_asynccnt 0" ::: "memory");
    }

    const uint32_t sbase = (uint32_t)(g & 1) * STAGE_F + tid * 4;
#pragma unroll
    for (int r = 0; r < ROWS_PER_G; ++r) {
      // Direct __shared__ indexing -> addrspace(3) ds_load_b128.
      const float4 v = *reinterpret_cast<const float4*>(&smem[sbase + r * DIM]);
      a0 = fmaf(a0, EMA_OMA, EMA_ALPHA * v.x);
      a1 = fmaf(a1, EMA_OMA, EMA_ALPHA * v.y);
      a2 = fmaf(a2, EMA_OMA, EMA_ALPHA * v.z);
      a3 = fmaf(a3, EMA_OMA, EMA_ALPHA * v.w);
    }
  }

  float4* o = reinterpret_cast<float4*>(ws) + (size_t)blockIdx.x * (DIM / 4) + tid;
  *o = make_float4(a0, a1, a2, a3);
}

// Pass 2: out[b,d] = sum_c scale[c] * ws[b,c,d],  scale[c] = 0.9^(64*(63-c)).
// One wave per (batch, 16-dim tile). WMMA f32 16x16x4: K = chunk index,
// N = dim tile, A = scale vector replicated across the 16 M-rows, C chained.
// A layout (32-bit 16x4): VGPR0 = K0 (lanes 0-15) / K2 (lanes 16-31),
//                         VGPR1 = K1 / K3.
// B layout (4x16, rows across lane halves): VGPR0 = K0/K1, VGPR1 = K2/K3.
__global__ __launch_bounds__(32) void ema_combine_wmma(
    const float* __restrict__ ws, float* __restrict__ out) {
  const int lane  = threadIdx.x;
  const int b     = blockIdx.x >> 6;          // DIM/16 = 64 tiles per batch
  const int dbase = (blockIdx.x & 63) * 16;
  const int half  = lane >> 4;
  const int n     = lane & 15;

  const float L2_09 = -0.15200309344504997f;  // log2(0.9)

  v8f c = {};
#pragma unroll
  for (int kg = 0; kg < NCHUNK / 4; ++kg) {
    const int c0 = kg * 4;
    const int ka = c0 + half * 2;             // this lane's A K-pair
    v2f A, B;
    A.x = exp2f(L2_09 * (float)(CHUNKLEN * (NCHUNK - 1 - ka)));
    A.y = exp2f(L2_09 * (float)(CHUNKLEN * (NCHUNK - 2 - ka)));
    B.x = ws[(size_t)(b * NCHUNK + c0 + half)     * DIM + dbase + n];
    B.y = ws[(size_t)(b * NCHUNK + c0 + 2 + half) * DIM + dbase + n];
    c = __builtin_amdgcn_wmma_f32_16x16x4_f32(
        /*neg_a=*/false, A, /*neg_b=*/false, B,
        /*c_mod=*/(short)0, c, /*reuse_a=*/false, /*reuse_b=*/false);
  }

  // All 16 D-rows are identical; row M=0 lives in D VGPR0, lanes 0-15.
  if (lane < 16) out[(size_t)b * DIM + dbase + lane] = c[0];
}

extern "C" void kernel_launch(void* const* d_in, const int* in_sizes, int n_in,
                              void* d_out, int out_size, void* d_ws, size_t ws_size,
                              hipStream_t stream) {
  const float* x = (const float*)d_in[0];
  float* out = (float*)d_out;
  float* ws  = (float*)d_ws;   // needs BATCH*NCHUNK*DIM*4 = 2 MiB

  ema_chunk_partial<<<BATCH * NCHUNK, 256, 0, stream>>>(x, ws);
  ema_combine_wmma<<<BATCH * (DIM / 16), 32, 0, stream>>>(ws, out);
}